// QuantumAttentionMechanism_27925877358862
// MI455X (gfx1250) — compile-verified
//
#include <hip/hip_runtime.h>
#include <hip/hip_bf16.h>

// ---------------------------------------------------------------------------
// QuantumAttentionMechanism on MI455X (gfx1250)
// Math collapses: weights == 1/2048 exactly (softmax of a constant);
//                 out[b,s,:] == ((mean_t feat[b]) @ Wv.T + bv) @ Wo.T + bo  (same for all s)
// Dominant cost: writing 537MB of constant weights -> NT B128 stores (537MB >> 192MB L2).
// Tiny 16x512x512 GEMMs done with V_WMMA_F32_16X16X4_F32, branch-free inner loop.
// ---------------------------------------------------------------------------

#define B_  4
#define S_  2048
#define F_  512
#define H_  8
#define MPAD 16   // WMMA M tile; rows B_..15 are zero padding

typedef __attribute__((ext_vector_type(2))) float v2f;
typedef __attribute__((ext_vector_type(4))) float v4f;
typedef __attribute__((ext_vector_type(8))) float v8f;

// ---- Stage 1a: partial column sums of features over sequence chunks -------
// grid: B_*16 blocks x 512 threads. partial[(b*16+chunk)*512 + f] = sum of 128 t's.
__global__ void qam_partial_mean(const float* __restrict__ feat,
                                 float* __restrict__ partial) {
  const int b     = blockIdx.x >> 4;
  const int chunk = blockIdx.x & 15;
  const int f     = threadIdx.x;
  const float* p = feat + ((size_t)b * S_ + (size_t)chunk * 128) * F_ + f;
  float s = 0.0f;
#pragma unroll 4
  for (int t = 0; t < 128; ++t) s += p[(size_t)t * F_];
  partial[((size_t)blockIdx.x << 9) + f] = s;
}

// ---- Stage 1b: reduce 16 partials -> meanFeat[16x512], rows >= B_ zeroed ---
// grid: (MPAD*F_)/256 = 32 blocks x 256 threads
__global__ void qam_reduce_mean(const float* __restrict__ partial,
                                float* __restrict__ meanFeatPad) {
  const int idx = blockIdx.x * blockDim.x + threadIdx.x;  // row*512 + f
  const int row = idx >> 9;
  const int f   = idx & 511;
  float s = 0.0f;
  if (row < B_) {
#pragma unroll
    for (int c = 0; c < 16; ++c) s += partial[((size_t)(row * 16 + c) << 9) + f];
    s *= (1.0f / (float)S_);
  }
  meanFeatPad[idx] = s;   // padding rows get exact 0
}

// ---- Stage 2: GEMM  O[m][n] = sum_k A[m][k]*W[n][k] + bias[n], M=16,N=K=512
// One wave per 16-col tile; 4 blocks x 256 threads = 32 waves = 512 columns.
// V_WMMA_F32_16X16X4_F32 VGPR layouts (ISA 7.12.2):
//   A (16x4):  lanes 0-15 -> M=lane, v[0]=K=k, v[1]=K=k+1 ; lanes 16-31 -> K=k+2,k+3
//   B (4x16):  lanes 0-15 -> N=lane, v[0]=K=k, v[1]=K=k+1 ; lanes 16-31 -> K=k+2,k+3
//   C/D:       lanes 0-15 -> N=lane, vgpr j -> M=j ; lanes 16-31 -> M=j+8
// A is the zero-padded 16x512 buffer -> all loads unconditional, EXEC all-1s.
__global__ void qam_gemm16_wmma(const float* __restrict__ A,
                                const float* __restrict__ W,
                                const float* __restrict__ bias,
                                float* __restrict__ O) {
  const int lane = threadIdx.x & 31;
  const int wave = (blockIdx.x * blockDim.x + threadIdx.x) >> 5;  // 0..31
  const int n0   = wave * 16;
  const int half = lane >> 4;     // 0: lanes 0-15, 1: lanes 16-31
  const int l    = lane & 15;

  const float* arow = A + (size_t)l * F_ + 2 * half;          // M = l
  const float* wrow = W + (size_t)(n0 + l) * F_ + 2 * half;   // N = n0 + l

  v8f c = {};
#pragma unroll 4
  for (int k = 0; k < F_; k += 4) {
    v2f a = *reinterpret_cast<const v2f*>(arow + k);   // global_load_b64
    v2f b = *reinterpret_cast<const v2f*>(wrow + k);   // global_load_b64
    c = __builtin_amdgcn_wmma_f32_16x16x4_f32(
        /*neg_a=*/false, a, /*neg_b=*/false, b,
        /*c_mod=*/(short)0, c, /*reuse_a=*/false, /*reuse_b=*/false);
  }

  // Branch-free store of all 16 rows (pad rows = bias, finite, never read).
  const float bn = bias[n0 + l];
  float* ocol = O + (size_t)(half * 8) * F_ + n0 + l;
#pragma unroll
  for (int j = 0; j < 8; ++j) ocol[(size_t)j * F_] = c[j] + bn;
}

// ---- Stage 3: broadcast o[b,:] to out[b,s,:] ------------------------------
// grid: (B*S*F/4)/256 blocks x 256 threads, B128 stores.
__global__ void qam_broadcast_out(const float* __restrict__ o,
                                  v4f* __restrict__ out) {
  const size_t i  = (size_t)blockIdx.x * blockDim.x + threadIdx.x;  // v4f idx
  const int f4 = (int)(i & 127);          // 128 v4f per row of 512
  const int bs = (int)(i >> 7);           // (b,s) pair
  const int b  = bs >> 11;                // / 2048
  out[i] = reinterpret_cast<const v4f*>(o)[b * 128 + f4];
}

// ---- Stage 4: fill weights with exactly 1/2048 ----------------------------
// 537MB >> 192MB L2 -> non-temporal stores to avoid trashing L2.
__global__ void qam_fill_weights(v4f* __restrict__ w, size_t n4) {
  const float v = 1.0f / (float)S_;       // exact: 0x3A000000
  const v4f val = {v, v, v, v};
  const size_t stride = (size_t)gridDim.x * blockDim.x;
  for (size_t i = (size_t)blockIdx.x * blockDim.x + threadIdx.x; i < n4; i += stride) {
    __builtin_nontemporal_store(val, &w[i]);
  }
}

extern "C" void kernel_launch(void* const* d_in, const int* in_sizes, int n_in,
                              void* d_out, int out_size, void* d_ws, size_t ws_size,
                              hipStream_t stream) {
  const float* features = (const float*)d_in[0];
  // d_in[1] navigation_state: unused by reference math
  // d_in[2..5] Wq,bq,Wk,bk: dead (Q/K overridden by quantum step)
  const float* Wv = (const float*)d_in[6];
  const float* bv = (const float*)d_in[7];
  // d_in[8] amplitudes, d_in[9] phases: only enter via softmax(const) == 1/S
  const float* Wo = (const float*)d_in[10];
  const float* bo = (const float*)d_in[11];

  float* out_main = (float*)d_out;                              // [B,S,F] = 4,194,304 f32
  float* weights  = out_main + (size_t)B_ * S_ * F_;            // [B,H,S,S] = 134,217,728 f32

  // workspace layout (floats)
  float* partial  = (float*)d_ws;                  // B*16*512 = 32768
  float* meanFeat = partial + B_ * 16 * F_;        // 16*512   =  8192 (zero-padded)
  float* mv       = meanFeat + MPAD * F_;          // 16*512   =  8192
  float* o        = mv + MPAD * F_;                // 16*512   =  8192

  // 1) per-batch column means of features (streams 16.8MB once)
  qam_partial_mean<<<B_ * 16, F_, 0, stream>>>(features, partial);
  qam_reduce_mean<<<(MPAD * F_) / 256, 256, 0, stream>>>(partial, meanFeat);

  // 2) mv = meanFeat @ Wv.T + bv ; o = mv @ Wo.T + bo   (WMMA f32 16x16x4)
  qam_gemm16_wmma<<<4, 256, 0, stream>>>(meanFeat, Wv, bv, mv);
  qam_gemm16_wmma<<<4, 256, 0, stream>>>(mv, Wo, bo, o);

  // 3) out[b,s,:] = o[b,:]
  {
    const size_t n4 = (size_t)B_ * S_ * F_ / 4;   // 1,048,576 v4f
    qam_broadcast_out<<<(unsigned)(n4 / 256), 256, 0, stream>>>(o, (v4f*)out_main);
  }

  // 4) weights = 1/2048 everywhere (537MB, NT stores)
  {
    const size_t n4 = (size_t)B_ * H_ * S_ * S_ / 4;  // 33,554,432 v4f
    qam_fill_weights<<<8192, 256, 0, stream>>>((v4f*)weights, n4);
  }
}